// GNN_49289044689459
// MI455X (gfx1250) — compile-verified
//
#include <hip/hip_runtime.h>
#include <hip/hip_bf16.h>

typedef __attribute__((ext_vector_type(2))) float v2f;
typedef __attribute__((ext_vector_type(8))) float v8f;

#define HDIM 128
#define KDIM 256

// ---------------------------------------------------------------------------
// Zero-fill, float4-wide (graph-capture-safe replacement for hipMemsetAsync)
// n must be a multiple of 4 (all our buffers are).
// ---------------------------------------------------------------------------
__global__ void gnn_zero4_kernel(float4* __restrict__ p, int n4) {
    int i = blockIdx.x * blockDim.x + threadIdx.x;
    if (i < n4) p[i] = make_float4(0.f, 0.f, 0.f, 0.f);
}

__global__ void gnn_zero_kernel(float* __restrict__ p, int n) {
    int i = blockIdx.x * blockDim.x + threadIdx.x;
    if (i < n) p[i] = 0.0f;
}

// ---------------------------------------------------------------------------
// Degree: deg[dst] += 1 per edge (computed once, reused by both layers)
// ---------------------------------------------------------------------------
__global__ void gnn_degree_kernel(const long long* __restrict__ dst,
                                  float* __restrict__ deg, int E) {
    int e = blockIdx.x * blockDim.x + threadIdx.x;
    if (e < E) atomicAdd(deg + (int)dst[e], 1.0f);
}

// ---------------------------------------------------------------------------
// Scatter: agg[dst] += x[src] * att[e].
// One wave32 per edge. The edge id is forced wave-uniform via readfirstlane
// so src/dst/att become SCALAR loads (s_load on KMcnt) instead of 32
// redundant vector loads, and the row gather / atomics use saddr+voffset
// forms. Lane L handles features [4L,4L+3]: one coalesced b128 load
// (512B per edge row) and 4 global_atomic_add_f32 into agg, which is
// L2-resident (25.6MB << 192MB L2), so atomics resolve in-cache.
// ---------------------------------------------------------------------------
__global__ __launch_bounds__(256)
void gnn_scatter_kernel(const float* __restrict__ X,
                        const long long* __restrict__ src,
                        const long long* __restrict__ dst,
                        const float* __restrict__ att,
                        float* __restrict__ agg, int E) {
    // 8 edges per 256-thread block; e is uniform across each wave32.
    int e = __builtin_amdgcn_readfirstlane(blockIdx.x * 8 + (threadIdx.x >> 5));
    int lane = threadIdx.x & 31;
    if (e >= E) return;
    int s = (int)src[e];           // scalar load
    int d = (int)dst[e];           // scalar load
    float w = att[e];              // scalar load
    const float4 xv = *(const float4*)(X + (size_t)s * HDIM + lane * 4);
    float* base = agg + (size_t)d * HDIM + lane * 4;
    atomicAdd(base + 0, xv.x * w);
    atomicAdd(base + 1, xv.y * w);
    atomicAdd(base + 2, xv.z * w);
    atomicAdd(base + 3, xv.w * w);
}

// ---------------------------------------------------------------------------
// Mean: agg[n][h] /= max(deg[n], 1)
// ---------------------------------------------------------------------------
__global__ void gnn_mean_kernel(float* __restrict__ agg,
                                const float* __restrict__ deg, int total) {
    int i = blockIdx.x * blockDim.x + threadIdx.x;
    if (i < total) {
        float d = deg[i >> 7];  // /128
        agg[i] *= 1.0f / fmaxf(d, 1.0f);
    }
}

// ---------------------------------------------------------------------------
// Fused concat + GEMM + bias + ReLU using V_WMMA_F32_16X16X4_F32.
//   Y[16n, 128] = relu( [X | AGG][16n, 256] @ W[256,128] + B )
// Block = 256 threads = 8 waves. Each block owns a 16-node row tile; the
// 16x256 fp32 A-tile is staged once in LDS (rows padded to 258 floats so
// the 16-row stride-258 column reads hit distinct banks; ds_load_2addr_b64
// per fragment). Each wave computes one 16x16 output tile via 64 chained
// v_wmma_f32_16x16x4_f32 with fp32 accumulation (
// this workload is scatter/bandwidth-bound so fp32 WMMA is free).
//
// WMMA wave32 layouts (ISA 7.12.2):
//   A 16x4 f32 : lane L -> row M=L&15; lanes<16 hold K={k0,k0+1},
//                lanes>=16 hold K={k0+2,k0+3}  (v2f per lane)
//   B 4x16 f32 : lane L -> col N=L&15; same K-half split (v2f per lane)
//   C/D 16x16  : VGPR v, lane L -> D[v + 8*(L>>4)][L&15]   (v8f)
// ---------------------------------------------------------------------------
__global__ __launch_bounds__(256)
void gnn_gemm_relu_kernel(const float* __restrict__ X,    // [N,128]
                          const float* __restrict__ AGG,  // [N,128] mean agg
                          const float* __restrict__ W,    // [256,128] row-major
                          const float* __restrict__ Bv,   // [128]
                          float* __restrict__ Y,          // [N,128]
                          int N) {
    __shared__ float As[16][KDIM + 2];  // pitch 258: stride%64=2 -> no bank clash

    const int row0 = blockIdx.x * 16;
    const int tid  = threadIdx.x;

    // Cooperative stage of the concatenated A tile: 16 rows x 256 cols.
    for (int i = tid; i < 16 * KDIM; i += 256) {
        int r = i >> 8;         // row in tile
        int k = i & (KDIM - 1); // 0..255
        int node = row0 + r;
        if (node >= N) node = N - 1;  // tail clamp (N=50000 is 16-aligned)
        float v = (k < HDIM) ? X[(size_t)node * HDIM + k]
                             : AGG[(size_t)node * HDIM + (k - HDIM)];
        As[r][k] = v;
    }
    __syncthreads();

    const int wave = tid >> 5;
    const int lane = tid & 31;
    const int half = lane >> 4;   // K-half select for A/B fragments
    const int n    = lane & 15;   // col within tile (B/C/D) == row M for A
    const int col  = wave * 16 + n;

    v8f c = {};
    #pragma unroll 8
    for (int k0 = 0; k0 < KDIM; k0 += 4) {
        const int ka = k0 + 2 * half;
        // A fragment: 8B-aligned LDS float2 (ka even, row pitch 258*4 B)
        v2f a = *(const v2f*)&As[n][ka];
        // B fragment: two dword loads of W column `col` (L2-resident, 131KB)
        v2f b;
        b.x = W[(size_t)ka * HDIM + col];
        b.y = W[(size_t)(ka + 1) * HDIM + col];
        // 8-arg form: (neg_a, A, neg_b, B, c_mod, C, reuse_a, reuse_b)
        c = __builtin_amdgcn_wmma_f32_16x16x4_f32(false, a, false, b,
                                                  (short)0, c, false, false);
    }

    // Epilogue: branch-free fast path for full tiles -> 8 coalesced
    // global_store_b32 at immediate offsets (v*512B) from one base.
    const float bias = Bv[col];
    float* yb = Y + (size_t)(row0 + 8 * half) * HDIM + col;
    if (row0 + 16 <= N) {
        #pragma unroll
        for (int v = 0; v < 8; ++v) {
            yb[(size_t)v * HDIM] = fmaxf(c[v] + bias, 0.0f);
        }
    } else {
        #pragma unroll
        for (int v = 0; v < 8; ++v) {
            if (row0 + 8 * half + v < N)
                yb[(size_t)v * HDIM] = fmaxf(c[v] + bias, 0.0f);
        }
    }
}

// ---------------------------------------------------------------------------
// Launch: deg -> [zero agg, scatter, mean, gemm] x 2 layers
// ---------------------------------------------------------------------------
extern "C" void kernel_launch(void* const* d_in, const int* in_sizes, int n_in,
                              void* d_out, int out_size, void* d_ws, size_t ws_size,
                              hipStream_t stream) {
    const float*     data = (const float*)d_in[0];
    const long long* edge = (const long long*)d_in[1];  // int64 [2,E]
    const float*     att  = (const float*)d_in[2];
    const float*     w1   = (const float*)d_in[3];
    const float*     b1   = (const float*)d_in[4];
    const float*     w2   = (const float*)d_in[5];
    const float*     b2   = (const float*)d_in[6];
    float*           out  = (float*)d_out;

    const int N = in_sizes[0] / HDIM;   // 50000
    const int E = in_sizes[2];          // 640000
    const long long* src = edge;        // edge[0]
    const long long* dst = edge + E;    // edge[1]

    // Scratch layout in d_ws
    char* ws = (char*)d_ws;
    size_t off = 0;
    auto alloc = [&](size_t bytes) {
        void* p = ws + off;
        off += (bytes + 255) & ~(size_t)255;
        return p;
    };
    float* deg = (float*)alloc((size_t)N * sizeof(float));
    float* agg = (float*)alloc((size_t)N * HDIM * sizeof(float));
    float* h1  = (float*)alloc((size_t)N * HDIM * sizeof(float));
    (void)ws_size; (void)n_in; (void)out_size;

    const int NH  = N * HDIM;
    const int zb4 = (NH / 4 + 255) / 256;
    const int mb  = (NH + 255) / 256;
    const int sb  = (E + 7) / 8;        // 8 edges (waves) per 256-thread block
    const int gb  = (N + 15) / 16;

    // Degree (shared by both layers)
    gnn_zero_kernel<<<(N + 255) / 256, 256, 0, stream>>>(deg, N);
    gnn_degree_kernel<<<(E + 255) / 256, 256, 0, stream>>>(dst, deg, E);

    // Layer 1: agg = mean_{e:dst=n} att*x[src] ; h1 = relu([x|agg] @ w1 + b1)
    gnn_zero4_kernel<<<zb4, 256, 0, stream>>>((float4*)agg, NH / 4);
    gnn_scatter_kernel<<<sb, 256, 0, stream>>>(data, src, dst, att, agg, E);
    gnn_mean_kernel<<<mb, 256, 0, stream>>>(agg, deg, NH);
    gnn_gemm_relu_kernel<<<gb, 256, 0, stream>>>(data, agg, w1, b1, h1, N);

    // Layer 2: same over h1 -> out
    gnn_zero4_kernel<<<zb4, 256, 0, stream>>>((float4*)agg, NH / 4);
    gnn_scatter_kernel<<<sb, 256, 0, stream>>>(h1, src, dst, att, agg, E);
    gnn_mean_kernel<<<mb, 256, 0, stream>>>(agg, deg, NH);
    gnn_gemm_relu_kernel<<<gb, 256, 0, stream>>>(h1, agg, w2, b2, out, N);
}